// MutiClassPostProcess_62319975465065
// MI455X (gfx1250) — compile-verified
//
#include <hip/hip_runtime.h>
#include <stdint.h>

#define BATCH   64
#define NCLASS  80
#define NOBJ    1000
#define TOPK    100
#define NPB     (NCLASS * NOBJ)      // 80000 logits per batch row
#define NBUCKET 2048
#define BSHIFT  21                   // 32 - 11 bits
#define THREADS 256
#define NHIST   8                    // one histogram copy per wave32
#define TILE_ELEMS 2048              // 8 KB per tile; 8 floats (2x b128) per lane
#define NTILES  ((NPB + TILE_ELEMS - 1) / TILE_ELEMS)   // 40 (last tile = 128 elems)
#define DEPTH   24                   // async ring depth: 192 KB in flight per block
#define CAP     1024                 // candidate cap (typical count ~110)

static_assert(DEPTH == 24, "s_wait_asynccnt immediates below assume DEPTH==24");
static_assert(2 * DEPTH <= 63, "ASYNCcnt is a 6-bit counter");
static_assert(NTILES > DEPTH, "pipeline assumes more tiles than ring depth");

// ---- CDNA5 async global->LDS DMA (ASYNCcnt-tracked, CDNA5 ISA) ----
__device__ __forceinline__ void async_copy_b128(void* lds_dst, const void* gsrc) {
  uint32_t lds_addr = (uint32_t)(uintptr_t)lds_dst;   // low 32 bits = LDS offset
  asm volatile("global_load_async_to_lds_b128 %0, %1, off"
               :: "v"(lds_addr), "v"(gsrc)
               : "memory");
}
// steady state: 2 ops/tile * DEPTH tiles outstanding = 48; <=46 => oldest tile done
__device__ __forceinline__ void wait_async_steady() {
  asm volatile("s_wait_asynccnt 46" ::: "memory");
}
__device__ __forceinline__ void wait_async_drain() {
  asm volatile("s_wait_asynccnt 0" ::: "memory");
}

__device__ __forceinline__ uint32_t order_key(float x) {
  uint32_t u = __float_as_uint(x);
  uint32_t mask = (uint32_t)((int32_t)u >> 31) | 0x80000000u;
  return u ^ mask;   // larger float -> larger key
}

// Deep ring-buffered streamer over one batch row of logits.
// fn(float4 v, int elemBase) is called for each aligned group of 4 elements.
template <typename F>
__device__ __forceinline__ void stream_logits(const float* __restrict__ src,
                                              float (*buf)[TILE_ELEMS],
                                              int tid, F fn) {
  // prologue: fill the ring
#pragma unroll
  for (int t = 0; t < DEPTH; ++t) {
    int e0 = t * TILE_ELEMS + tid * 8;
    if (e0 < NPB)     async_copy_b128(&buf[t][tid * 8],     src + e0);
    if (e0 + 4 < NPB) async_copy_b128(&buf[t][tid * 8 + 4], src + e0 + 4);
  }
  int t = 0;
  for (; t < NTILES - DEPTH; ++t) {
    wait_async_steady();      // oldest tile's 2 ops retired for this wave
    __syncthreads();          // all waves' portions of tile t visible
    const int slot = t % DEPTH;
    const int e0 = t * TILE_ELEMS + tid * 8;
    {
      float4 a = *(const float4*)&buf[slot][tid * 8];
      float4 c = *(const float4*)&buf[slot][tid * 8 + 4];
      fn(a, e0);
      fn(c, e0 + 4);
    }
    __syncthreads();          // everyone done reading slot before overwrite
    const int n0 = (t + DEPTH) * TILE_ELEMS + tid * 8;
    if (n0 < NPB)     async_copy_b128(&buf[slot][tid * 8],     src + n0);
    if (n0 + 4 < NPB) async_copy_b128(&buf[slot][tid * 8 + 4], src + n0 + 4);
  }
  wait_async_drain();         // remaining <= DEPTH tiles all landed
  __syncthreads();
  for (; t < NTILES; ++t) {   // distinct slots: no further syncs needed
    const int slot = t % DEPTH;
    const int e0 = t * TILE_ELEMS + tid * 8;
    if (e0 < NPB) {
      float4 a = *(const float4*)&buf[slot][tid * 8];
      fn(a, e0);
      if (e0 + 4 < NPB) {
        float4 c = *(const float4*)&buf[slot][tid * 8 + 4];
        fn(c, e0 + 4);
      }
    }
  }
  __syncthreads();
}

__global__ __launch_bounds__(THREADS)
void topk_post_kernel(const float* __restrict__ target_sizes,
                      const float* __restrict__ pred_logits,
                      const float* __restrict__ pred_boxes,
                      float* __restrict__ out) {
  __shared__ __align__(16) float buf[DEPTH][TILE_ELEMS];   // 192 KB async ring
  __shared__ uint32_t hist8[NHIST][NBUCKET + 1];           // 65.6 KB, bank-skewed
  __shared__ uint32_t chunkSum[THREADS];                   // 1 KB
  __shared__ uint32_t candKey[CAP];                        // 4 KB
  __shared__ uint32_t candIdx[CAP];                        // 4 KB
  __shared__ uint32_t sh_thresh, sh_count;

  const int tid = threadIdx.x;
  const int hid = tid >> 5;                                // wave32 id
  const int b   = blockIdx.x;
  const float* src = pred_logits + (size_t)b * NPB;

  for (int i = tid; i < NHIST * (NBUCKET + 1); i += THREADS)
    ((uint32_t*)hist8)[i] = 0;
  if (tid == 0) sh_count = 0;
  __syncthreads();

  // -------- Pass 1 (HBM): radix histogram of order keys --------
  stream_logits(src, buf, tid, [&](float4 v, int /*base*/) {
    atomicAdd(&hist8[hid][order_key(v.x) >> BSHIFT], 1u);
    atomicAdd(&hist8[hid][order_key(v.y) >> BSHIFT], 1u);
    atomicAdd(&hist8[hid][order_key(v.z) >> BSHIFT], 1u);
    atomicAdd(&hist8[hid][order_key(v.w) >> BSHIFT], 1u);
  });

  // -------- Threshold scan: cumulative count from top bucket down --------
  {
    uint32_t s = 0;
#pragma unroll
    for (int j = 0; j < 8; ++j) {
      int bkt = tid * 8 + j;
      uint32_t c = 0;
#pragma unroll
      for (int w = 0; w < NHIST; ++w) c += hist8[w][bkt];
      s += c;
    }
    chunkSum[tid] = s;
    __syncthreads();
    if (tid == 0) {
      uint32_t cum = 0; int tc = 0;
      for (int i = THREADS - 1; i >= 0; --i) {
        uint32_t n = cum + chunkSum[i];
        if (n >= TOPK) { tc = i; break; }
        cum = n;
      }
      uint32_t thresh = (uint32_t)(tc * 8);
      for (int j = 7; j >= 0; --j) {
        int bkt = tc * 8 + j;
        uint32_t c = 0;
#pragma unroll
        for (int w = 0; w < NHIST; ++w) c += hist8[w][bkt];
        uint32_t n = cum + c;
        if (n >= TOPK) { thresh = (uint32_t)bkt; break; }
        cum = n;
      }
      sh_thresh = thresh;
    }
    __syncthreads();
  }
  const uint32_t thresh = sh_thresh;

  // -------- Pass 2 (L2-resident): collect candidates in buckets >= thresh --------
  stream_logits(src, buf, tid, [&](float4 v, int base) {
    float xs[4] = {v.x, v.y, v.z, v.w};
#pragma unroll
    for (int k = 0; k < 4; ++k) {
      uint32_t key = order_key(xs[k]);
      if ((key >> BSHIFT) >= thresh) {
        uint32_t pos = atomicAdd(&sh_count, 1u);
        if (pos < CAP) { candKey[pos] = key; candIdx[pos] = (uint32_t)(base + k); }
      }
    }
  });

  // -------- Exact rank among candidates (ties: lower index first, like lax.top_k) --------
  {
    uint32_t cnt = sh_count;
    int M = (int)(cnt < (uint32_t)CAP ? cnt : (uint32_t)CAP);
    const float img_h = target_sizes[2 * b + 0];
    const float img_w = target_sizes[2 * b + 1];
    for (int i = tid; i < M; i += THREADS) {
      uint32_t ki = candKey[i], xi = candIdx[i];
      int rank = 0;
      for (int j = 0; j < M; ++j) {
        uint32_t kj = candKey[j], xj = candIdx[j];
        rank += (int)((kj > ki) || (kj == ki && xj < xi));
      }
      if (rank < TOPK) {
        uint32_t ub = (ki & 0x80000000u) ? (ki ^ 0x80000000u) : ~ki;  // key -> bits
        float logit = __uint_as_float(ub);
        float score = 1.0f / (1.0f + expf(-logit));
        int cls = (int)(xi / NOBJ);
        int obj = (int)xi - cls * NOBJ;
        size_t boff = ((size_t)(b * NCLASS + cls) * NOBJ + (size_t)obj) * 4;
        float4 bx = *(const float4*)(pred_boxes + boff);
        float hw = 0.5f * bx.z, hh = 0.5f * bx.w;
        float4 o4;
        o4.x = (bx.x - hw) * img_w;
        o4.y = (bx.y - hh) * img_h;
        o4.z = (bx.x + hw) * img_w;
        o4.w = (bx.y + hh) * img_h;
        int o = b * TOPK + rank;
        out[o] = score;                                  // top_scores
        out[BATCH * TOPK + o] = (float)cls;              // top_labels
        *(float4*)(out + 2 * BATCH * TOPK + (size_t)o * 4) = o4;  // top_boxes
      }
    }
  }
}

extern "C" void kernel_launch(void* const* d_in, const int* in_sizes, int n_in,
                              void* d_out, int out_size, void* d_ws, size_t ws_size,
                              hipStream_t stream) {
  const float* target_sizes = (const float*)d_in[0];
  // d_in[1] = batch_index, d_in[2] = class_index (derivable analytically; unused)
  const float* pred_logits  = (const float*)d_in[3];
  const float* pred_boxes   = (const float*)d_in[4];
  float* out = (float*)d_out;
  topk_post_kernel<<<dim3(BATCH), dim3(THREADS), 0, stream>>>(
      target_sizes, pred_logits, pred_boxes, out);
}